// GraphConvolutionLayer_16054587753020
// MI455X (gfx1250) — compile-verified
//
#include <hip/hip_runtime.h>
#include <hip/hip_bf16.h>

typedef __attribute__((ext_vector_type(2))) float v2f;
typedef __attribute__((ext_vector_type(8))) float v8f;

#define F_IN  64
#define F_HID 64
#define F_OUT 32

// ---------------------------------------------------------------------------
// guaranteed single-instruction device-scope f32 atomic add (gfx1250).
// Non-returning form (tracked with STOREcnt; s_endpgm implicitly waits idle).
// scope:SCOPE_DEV is required: accumulation crosses workgroups, so the RMW
// must happen at the device (L2) level, not the default WGP scope.
// ---------------------------------------------------------------------------
__device__ __forceinline__ void atomic_add_f32(float* p, float v) {
    asm volatile("global_atomic_add_f32 %0, %1, off scope:SCOPE_DEV"
                 :: "v"(p), "v"(v)
                 : "memory");
}

// ---------------------------------------------------------------------------
// utility: fill a float buffer with a constant
// ---------------------------------------------------------------------------
__global__ void fill_kernel(float* __restrict__ p, float v, long long n) {
    long long i = (long long)blockIdx.x * blockDim.x + threadIdx.x;
    if (i < n) p[i] = v;
}

// ---------------------------------------------------------------------------
// degree accumulation: deg[dst[e]] += 1  (deg pre-filled with 1.0 = self loop)
// ---------------------------------------------------------------------------
__global__ void edge_degree_kernel(const long long* __restrict__ dst,
                                   float* __restrict__ deg, int E) {
    int e = blockIdx.x * blockDim.x + threadIdx.x;
    if (e < E) atomic_add_f32(&deg[(int)dst[e]], 1.0f);
}

// deg -> dinv = rsqrt(deg) in place (deg >= 1 always here)
__global__ void rsqrt_kernel(float* __restrict__ d, int n) {
    int i = blockIdx.x * blockDim.x + threadIdx.x;
    if (i < n) {
        float v = d[i];
        d[i] = (v > 0.0f) ? rsqrtf(v) : 0.0f;
    }
}

// ---------------------------------------------------------------------------
// H = X @ W  via V_WMMA_F32_16X16X4_F32.  X: [n x 64] row-major, W: [64 x NCOLS].
// One wave computes a 16-row strip across all NCOLS columns.
// A(16x4 f32) layout: lanes 0-15 -> K={0,1}, lanes 16-31 -> K={2,3}.
// C/D(16x16 f32): VGPR v, lanes 0-15 -> row v, lanes 16-31 -> row v+8; col = lane&15.
// ---------------------------------------------------------------------------
template <int NCOLS>
__global__ void gemm_wmma_kernel(const float* __restrict__ X,
                                 const float* __restrict__ W,
                                 float* __restrict__ H, int n_rows) {
    constexpr int NT = NCOLS / 16;
    int wave   = (int)((blockIdx.x * blockDim.x + threadIdx.x) >> 5);
    int strips = (n_rows + 15) >> 4;
    if (wave >= strips) return;                 // wave-uniform: EXEC stays all-1s

    int lane = threadIdx.x & 31;
    int lh   = lane >> 4;                       // 0: K pair {0,1}, 1: K pair {2,3}
    int l16  = lane & 15;
    int row0 = wave << 4;

    int arow = row0 + l16;
    if (arow > n_rows - 1) arow = n_rows - 1;   // clamp (tail safety; N%16==0 here)

    v8f acc[NT] = {};

#pragma unroll
    for (int kt = 0; kt < F_IN / 4; ++kt) {
        int kb = kt * 4 + lh * 2;
        v2f a = *(const v2f*)(X + (long long)arow * F_IN + kb);
#pragma unroll
        for (int t = 0; t < NT; ++t) {
            v2f b;
            b.x = W[(kb + 0) * NCOLS + t * 16 + l16];
            b.y = W[(kb + 1) * NCOLS + t * 16 + l16];
            acc[t] = __builtin_amdgcn_wmma_f32_16x16x4_f32(
                false, a, false, b, (short)0, acc[t], false, false);
        }
    }

#pragma unroll
    for (int t = 0; t < NT; ++t) {
#pragma unroll
        for (int v = 0; v < 8; ++v) {
            int row = row0 + v + lh * 8;
            if (row < n_rows)
                H[(long long)row * NCOLS + t * 16 + l16] = acc[t][v];
        }
    }
}

// ---------------------------------------------------------------------------
// Layer-1 scatter: one wave per edge, each lane handles 2 contiguous features.
// agg[dst] += h[src] * dinv[src] * dinv[dst]
// Edge index is wave-uniform: readfirstlane lets src/dst/dinv become s_loads.
// ---------------------------------------------------------------------------
__global__ void scatter64_kernel(const long long* __restrict__ src,
                                 const long long* __restrict__ dst,
                                 const float* __restrict__ dinv,
                                 const float* __restrict__ h,
                                 float* __restrict__ agg, int E) {
    int e = blockIdx.x * (blockDim.x >> 5) + (threadIdx.x >> 5);
    if (e >= E) return;                          // wave-uniform branch
    e = __builtin_amdgcn_readfirstlane(e);       // prove uniformity -> s_load path
    int lane = threadIdx.x & 31;
    int s = (int)src[e];
    int d = (int)dst[e];
    float nrm = dinv[s] * dinv[d];
    const float2 hv = *(const float2*)(h + (long long)s * F_HID + lane * 2);
    float* base = agg + (long long)d * F_HID + lane * 2;
    atomic_add_f32(base + 0, hv.x * nrm);
    atomic_add_f32(base + 1, hv.y * nrm);
}

// ---------------------------------------------------------------------------
// Layer-2 scatter: one wave per edge, one feature per lane (F_OUT == 32).
// ---------------------------------------------------------------------------
__global__ void scatter32_kernel(const long long* __restrict__ src,
                                 const long long* __restrict__ dst,
                                 const float* __restrict__ dinv,
                                 const float* __restrict__ h,
                                 float* __restrict__ agg, int E) {
    int e = blockIdx.x * (blockDim.x >> 5) + (threadIdx.x >> 5);
    if (e >= E) return;                          // wave-uniform branch
    e = __builtin_amdgcn_readfirstlane(e);
    int lane = threadIdx.x & 31;
    int s = (int)src[e];
    int d = (int)dst[e];
    float nrm = dinv[s] * dinv[d];
    float v = h[(long long)s * F_OUT + lane] * nrm;
    atomic_add_f32(agg + (long long)d * F_OUT + lane, v);
}

// ---------------------------------------------------------------------------
// h1 <- relu(agg1 + h1 * dinv[i]^2 + b1[f])   (self-loop + bias + relu, in place)
// ---------------------------------------------------------------------------
__global__ void bias_relu_kernel(const float* __restrict__ agg,
                                 float* __restrict__ h,
                                 const float* __restrict__ dinv,
                                 const float* __restrict__ b, long long total) {
    long long idx = (long long)blockIdx.x * blockDim.x + threadIdx.x;
    if (idx >= total) return;
    int i = (int)(idx >> 6);                 // node
    int f = (int)(idx & 63);                 // feature
    float s = dinv[i];
    float v = agg[idx] + h[idx] * s * s + b[f];
    h[idx] = v > 0.0f ? v : 0.0f;
}

// ---------------------------------------------------------------------------
// out <- out + h2 * dinv[i]^2 + b2[f]   (self-loop + bias for layer 2)
// ---------------------------------------------------------------------------
__global__ void final_kernel(float* __restrict__ out,
                             const float* __restrict__ h2,
                             const float* __restrict__ dinv,
                             const float* __restrict__ b, long long total) {
    long long idx = (long long)blockIdx.x * blockDim.x + threadIdx.x;
    if (idx >= total) return;
    int i = (int)(idx >> 5);
    int f = (int)(idx & 31);
    float s = dinv[i];
    out[idx] = out[idx] + h2[idx] * s * s + b[f];
}

// ---------------------------------------------------------------------------
extern "C" void kernel_launch(void* const* d_in, const int* in_sizes, int n_in,
                              void* d_out, int out_size, void* d_ws, size_t ws_size,
                              hipStream_t stream) {
    const float*     x    = (const float*)d_in[0];
    const long long* eidx = (const long long*)d_in[1];   // int64 per reference
    const float*     W1   = (const float*)d_in[2];
    const float*     b1   = (const float*)d_in[3];
    const float*     W2   = (const float*)d_in[4];
    const float*     b2   = (const float*)d_in[5];
    float*           out  = (float*)d_out;

    const int N = in_sizes[0] / F_IN;
    const int E = in_sizes[1] / 2;
    const long long* esrc = eidx;
    const long long* edst = eidx + E;

    // workspace layout (floats): dinv[N] | h1[N*64] | agg1[N*64] | h2[N*32]
    float* dinv = (float*)d_ws;
    float* h1   = dinv + N;
    float* agg1 = h1 + (size_t)N * F_HID;
    float* h2   = agg1 + (size_t)N * F_HID;

    const int T = 256;
    long long nh = (long long)N * F_HID;
    long long no = (long long)N * F_OUT;

    // 0) init: deg=1 (self loop), agg1=0, out=0
    fill_kernel<<<(int)((N + T - 1) / T), T, 0, stream>>>(dinv, 1.0f, N);
    fill_kernel<<<(int)((nh + T - 1) / T), T, 0, stream>>>(agg1, 0.0f, nh);
    fill_kernel<<<(int)((no + T - 1) / T), T, 0, stream>>>(out, 0.0f, no);

    // 1) degrees -> dinv = rsqrt(deg)
    edge_degree_kernel<<<(E + T - 1) / T, T, 0, stream>>>(edst, dinv, E);
    rsqrt_kernel<<<(N + T - 1) / T, T, 0, stream>>>(dinv, N);

    // 2) h1 = x @ W1   (WMMA f32, one wave per 16-row strip, 8 waves/block)
    int strips = (N + 15) / 16;
    int gblocks = (strips + 7) / 8;
    gemm_wmma_kernel<F_HID><<<gblocks, T, 0, stream>>>(x, W1, h1, N);

    // 3) edge scatter for layer 1 (8 edges per 256-thread block)
    scatter64_kernel<<<(E + 7) / 8, T, 0, stream>>>(esrc, edst, dinv, h1, agg1, E);

    // 4) self-loop + bias + relu, h1 <- relu(...)
    bias_relu_kernel<<<(int)((nh + T - 1) / T), T, 0, stream>>>(agg1, h1, dinv, b1, nh);

    // 5) h2 = h1 @ W2
    gemm_wmma_kernel<F_OUT><<<gblocks, T, 0, stream>>>(h1, W2, h2, N);

    // 6) edge scatter for layer 2 into d_out
    scatter32_kernel<<<(E + 7) / 8, T, 0, stream>>>(esrc, edst, dinv, h2, out, E);

    // 7) self-loop + bias for layer 2
    final_kernel<<<(int)((no + T - 1) / T), T, 0, stream>>>(out, h2, dinv, b2, no);
}